// CollectDiffuseAttention_28123445854928
// MI455X (gfx1250) — compile-verified
//
#include <hip/hip_runtime.h>
#include <hip/hip_bf16.h>
#include <math.h>

// Problem constants (from reference)
constexpr int BB = 256;
constexpr int LL = 1024;
constexpr int DD = 512;
constexpr float INV_T = 0.04419417382415922f; // 1/sqrt(512)

typedef __attribute__((ext_vector_type(2))) float v2f;
typedef __attribute__((ext_vector_type(8))) float v8f;

// ---------------------------------------------------------------------------
// Kernel 1: per-batch collect logits (WMMA), gate logits (WMMA), softmax,
//           attn = softmax(logits) @ V.  One workgroup (8 wave32) per batch.
// ---------------------------------------------------------------------------
__global__ __launch_bounds__(256) void cda_phase1(
    const float* __restrict__ q,   // (B, D)
    const float* __restrict__ kc,  // (B, L, D)
    const float* __restrict__ kd,  // (B, L, D)
    const float* __restrict__ v,   // (B, L, D)
    float* __restrict__ logits_out, // (B, L)
    float* __restrict__ attn_ws,    // (B, D)
    float* __restrict__ gate_ws)    // (B, L)
{
    const int b    = blockIdx.x;
    const int t    = threadIdx.x;
    const int lane = t & 31;
    const int wave = t >> 5;
    const int half = (lane >> 4) & 1;  // K-pair select for f32 WMMA fragments
    const int m    = lane & 15;        // A-matrix row within tile

    __shared__ float q_s[DD];      // query
    __shared__ float p_s[LL];      // collect logits, then softmax probabilities
    __shared__ float g_s[LL];      // raw diffuse (gate) logits
    __shared__ float red_s[256];   // reduction scratch

    // ---- load q into LDS (coalesced) ----
    q_s[t]       = q[(size_t)b * DD + t];
    q_s[t + 256] = q[(size_t)b * DD + t + 256];
    __syncthreads();

    const size_t base = (size_t)b * LL * DD;

    // ---- Phase B: logits & gate logits via V_WMMA_F32_16X16X4_F32 ----
    // Each wave handles 16-row tiles: tiles (wave, wave+8, ..., wave+56).
    for (int tile = wave; tile < (LL / 16); tile += 8) {
        const int row = tile * 16 + m;
        const float* kcp = kc + base + (size_t)row * DD + 2 * half;
        const float* kdp = kd + base + (size_t)row * DD + 2 * half;

        v8f acc_c = {};
        v8f acc_d = {};
        // B fragment: q broadcast into all 16 columns.
        // lanes 0-15 need q[k],q[k+1]; lanes 16-31 need q[k+2],q[k+3].
        #pragma unroll 4
        for (int k = 0; k < DD; k += 4) {
            v2f a_c = *(const v2f*)(kcp + k);
            v2f a_d = *(const v2f*)(kdp + k);
            v2f bq  = *(const v2f*)(&q_s[k + 2 * half]);
            acc_c = __builtin_amdgcn_wmma_f32_16x16x4_f32(
                false, a_c, false, bq, (short)0, acc_c, false, false);
            acc_d = __builtin_amdgcn_wmma_f32_16x16x4_f32(
                false, a_d, false, bq, (short)0, acc_d, false, false);
        }
        // All 16 C columns are identical. Lane 0 holds rows tile*16+0..7 in
        // its 8 acc VGPRs; lane 16 holds rows tile*16+8..15.
        // Store RAW logits only; transcendentals happen later with full
        // thread parallelism instead of in this 2-lane divergent tail.
        if (m == 0) {
            #pragma unroll
            for (int r = 0; r < 8; ++r) {
                const int l = tile * 16 + half * 8 + r;
                p_s[l] = acc_c[r] * INV_T;
                g_s[l] = acc_d[r] * INV_T;
            }
        }
    }
    __syncthreads();

    // ---- write logits + gates to global (coalesced), local max ----
    float lmax = -3.402823466e38f;
    #pragma unroll
    for (int i = 0; i < 4; ++i) {
        const int l = t + 256 * i;
        const float lg = p_s[l];
        const float gd = g_s[l];
        // sigmoid via v_exp + v_rcp (approx rcp is fine for a gate)
        const float sg = __builtin_amdgcn_rcpf(1.0f + __expf(-gd));
        logits_out[(size_t)b * LL + l] = lg;
        gate_ws[(size_t)b * LL + l]   = sg;
        lmax = fmaxf(lmax, lg);
    }
    red_s[t] = lmax;
    __syncthreads();
    for (int off = 128; off > 0; off >>= 1) {
        if (t < off) red_s[t] = fmaxf(red_s[t], red_s[t + off]);
        __syncthreads();
    }
    const float mx = red_s[0];
    __syncthreads();

    // ---- exp + sum ----
    float lsum = 0.0f;
    #pragma unroll
    for (int i = 0; i < 4; ++i) {
        const int l = t + 256 * i;
        const float p = __expf(p_s[l] - mx);
        p_s[l] = p;
        lsum += p;
    }
    red_s[t] = lsum;
    __syncthreads();
    for (int off = 128; off > 0; off >>= 1) {
        if (t < off) red_s[t] = red_s[t] + red_s[t + off];
        __syncthreads();
    }
    const float inv_sum = 1.0f / red_s[0];
    __syncthreads();

    // ---- Phase C: attn[d] = inv_sum * sum_l p[l] * v[b,l,d] ----
    // Whole block reads each 2KB row of v coalesced; thread t owns d=2t,2t+1.
    const v2f* vrow = (const v2f*)(v + base);
    v2f acc = {};
    for (int l0 = 0; l0 < LL; l0 += 8) {
        if (l0 + 8 < LL) {
            __builtin_prefetch((const void*)(vrow + (size_t)(l0 + 8) * (DD / 2) + t), 0, 3);
        }
        #pragma unroll
        for (int u = 0; u < 8; ++u) {
            const int l = l0 + u;
            const float p = p_s[l];
            const v2f vv = __builtin_nontemporal_load(vrow + (size_t)l * (DD / 2) + t);
            acc.x += p * vv.x;
            acc.y += p * vv.y;
        }
    }
    acc.x *= inv_sum;
    acc.y *= inv_sum;
    ((v2f*)attn_ws)[(size_t)b * (DD / 2) + t] = acc;
}

// ---------------------------------------------------------------------------
// Kernel 2: out[b,l,d] = gate[b,l] * attn[b,d]  (pure store-bandwidth stream)
// grid = (L/64, B), 256 threads; each block writes 64 rows of 2KB.
// ---------------------------------------------------------------------------
__global__ __launch_bounds__(256) void cda_phase2(
    const float* __restrict__ gate_ws,  // (B, L)
    const float* __restrict__ attn_ws,  // (B, D)
    float* __restrict__ out)            // (B, L, D)
{
    const int b  = blockIdx.y;
    const int l0 = blockIdx.x * 64;
    const int t  = threadIdx.x;

    __shared__ float gsh[64];
    if (t < 64) gsh[t] = gate_ws[(size_t)b * LL + l0 + t];
    const v2f av = ((const v2f*)attn_ws)[(size_t)b * (DD / 2) + t];
    __syncthreads();

    v2f* out2 = (v2f*)out;
    const size_t rowbase = ((size_t)b * LL + l0) * (size_t)(DD / 2) + t;
    #pragma unroll 4
    for (int r = 0; r < 64; ++r) {
        const float g = gsh[r];
        v2f o;
        o.x = g * av.x;
        o.y = g * av.y;
        __builtin_nontemporal_store(o, &out2[rowbase + (size_t)r * (DD / 2)]);
    }
}

// ---------------------------------------------------------------------------
extern "C" void kernel_launch(void* const* d_in, const int* in_sizes, int n_in,
                              void* d_out, int out_size, void* d_ws, size_t ws_size,
                              hipStream_t stream) {
    const float* q  = (const float*)d_in[0];
    const float* kc = (const float*)d_in[1];
    const float* kd = (const float*)d_in[2];
    const float* v  = (const float*)d_in[3];

    float* out        = (float*)d_out;
    float* logits_out = out + (size_t)BB * LL * DD;  // tuple output #2

    float* attn_ws = (float*)d_ws;             // B*D floats
    float* gate_ws = attn_ws + (size_t)BB * DD; // B*L floats

    cda_phase1<<<dim3(BB), dim3(256), 0, stream>>>(q, kc, kd, v,
                                                   logits_out, attn_ws, gate_ws);
    cda_phase2<<<dim3(LL / 64, BB), dim3(256), 0, stream>>>(gate_ws, attn_ws, out);
}